// DecoderWithAttention_4904852652610
// MI455X (gfx1250) — compile-verified
//
#include <hip/hip_runtime.h>
#include <math.h>

// Problem constants (match reference)
#define Bc 256
#define Sc 512
#define Ec 512
#define Dc 512
#define Lc 64
#define Tc 64
#define Hc 63
#define G4 (4 * Dc) // 2048

typedef float v2f __attribute__((ext_vector_type(2)));
typedef float v8f __attribute__((ext_vector_type(8)));

// ---------------------------------------------------------------------------
// WMMA f32 16x16xK accumulation helper.
// A must point at A[m0][0] (row-major, leading dim lda).
// Normal B: points at B[0][n0] (row-major K x N, leading dim ldb).
// TRANSB:   points at Bt[n0][0] (row-major N x K, leading dim ldb) and computes
//           sum_k A[m][k] * Bt[n][k].
// Lane mapping per CDNA5 ISA 7.12.2:
//   A frag: lanes 0-15 -> (M=l, K=k0+{0,1}); lanes 16-31 -> (M=l, K=k0+{2,3})
//   B frag: lanes 0-15 -> (K=k0+{0,1}, N=l); lanes 16-31 -> (K=k0+{2,3}, N=l)
//   D frag: VGPR r -> rows m0+r (lanes 0-15) and m0+r+8 (lanes 16-31), col n0+l
// ---------------------------------------------------------------------------
template <bool TRANSB>
__device__ inline v8f wmma_span_f32(const float* __restrict__ A, int lda,
                                    const float* __restrict__ B, int ldb,
                                    int K, int l, int half, v8f acc) {
  const float* Ap = A + (size_t)l * lda + half * 2;
  if (TRANSB) {
    const float* Bp = B + (size_t)l * ldb + half * 2;
    for (int k0 = 0; k0 < K; k0 += 4) {
      v2f a, b;
      a.x = Ap[k0];
      a.y = Ap[k0 + 1];
      b.x = Bp[k0];
      b.y = Bp[k0 + 1];
      acc = __builtin_amdgcn_wmma_f32_16x16x4_f32(false, a, false, b, (short)0,
                                                  acc, false, false);
    }
  } else {
    for (int k0 = 0; k0 < K; k0 += 4) {
      v2f a, b;
      a.x = Ap[k0];
      a.y = Ap[k0 + 1];
      const float* Bp = B + (size_t)(k0 + half * 2) * ldb + l;
      b.x = Bp[0];
      b.y = Bp[ldb];
      acc = __builtin_amdgcn_wmma_f32_16x16x4_f32(false, a, false, b, (short)0,
                                                  acc, false, false);
    }
  }
  return acc;
}

// Generic GEMM: C[M x N] = epi(A[M x K] @ B + bias), tile 16x16 per wave,
// block = 256 threads = 8 waves covering 128 columns. grid = (M/16, N/128).
// EPI: 0 = none, 1 = tanh
template <bool TRANSB, int EPI>
__global__ __launch_bounds__(256) void gemm_wmma_kernel(
    const float* __restrict__ A, int lda, const float* __restrict__ B, int ldb,
    const float* __restrict__ bias, float* __restrict__ C, int ldc, int K) {
  const int lane = threadIdx.x & 31;
  const int wave = threadIdx.x >> 5;
  const int half = lane >> 4;
  const int l = lane & 15;
  const int m0 = blockIdx.x * 16;
  const int n0 = blockIdx.y * 128 + wave * 16;

  v8f acc;
  const float bv = bias ? bias[n0 + l] : 0.0f;
#pragma unroll
  for (int r = 0; r < 8; ++r) acc[r] = bv;

  const float* Bp = TRANSB ? (B + (size_t)n0 * ldb) : (B + n0);
  acc = wmma_span_f32<TRANSB>(A + (size_t)m0 * lda, lda, Bp, ldb, K, l, half, acc);

  float* Cp = C + (size_t)(m0 + half * 8) * ldc + n0 + l;
#pragma unroll
  for (int r = 0; r < 8; ++r) {
    float v = acc[r];
    if (EPI == 1) v = tanhf(v);
    Cp[(size_t)r * ldc] = v;
  }
}

// gates = [x_t | h] @ [W_ih ; W_hh] + b_ih + b_hh  -> (B x 2048)
// grid = (256/16, 2048/128) = (16,16), block 256
__global__ __launch_bounds__(256) void gates_wmma_kernel(
    const float* __restrict__ xt,    // B x 64
    const float* __restrict__ h,     // B x 512
    const float* __restrict__ W_ih,  // 64 x 2048
    const float* __restrict__ W_hh,  // 512 x 2048
    const float* __restrict__ b_ih, const float* __restrict__ b_hh,
    float* __restrict__ gates) {
  const int lane = threadIdx.x & 31;
  const int wave = threadIdx.x >> 5;
  const int half = lane >> 4;
  const int l = lane & 15;
  const int m0 = blockIdx.x * 16;
  const int n0 = blockIdx.y * 128 + wave * 16;

  v8f acc;
  const float bv = b_ih[n0 + l] + b_hh[n0 + l];
#pragma unroll
  for (int r = 0; r < 8; ++r) acc[r] = bv;

  acc = wmma_span_f32<false>(xt + (size_t)m0 * Lc, Lc, W_ih + n0, G4, Lc, l, half, acc);
  acc = wmma_span_f32<false>(h + (size_t)m0 * Dc, Dc, W_hh + n0, G4, Dc, l, half, acc);

  float* Cp = gates + (size_t)(m0 + half * 8) * G4 + n0 + l;
#pragma unroll
  for (int r = 0; r < 8; ++r) Cp[(size_t)r * G4] = acc[r];
}

// -------------------------- setup kernels ----------------------------------
__global__ void mean_enc_kernel(const float* __restrict__ cEnc,
                                float* __restrict__ meanEnc) {
  int idx = blockIdx.x * blockDim.x + threadIdx.x;  // over B*E
  if (idx >= Bc * Ec) return;
  int b = idx / Ec;
  int e = idx - b * Ec;
  const float* p = cEnc + (size_t)b * Sc * Ec + e;
  float s = 0.0f;
  for (int j = 0; j < Sc; ++j) s += p[(size_t)j * Ec];
  meanEnc[idx] = s * (1.0f / (float)Sc);
}

// carStates[t][b][l] = full[b, t+l], full = [hist(63) | cur(64)]
__global__ void car_states_kernel(const float* __restrict__ hist,
                                  const float* __restrict__ cur,
                                  float* __restrict__ cs) {
  int idx = blockIdx.x * blockDim.x + threadIdx.x;  // over T*B*L
  if (idx >= Tc * Bc * Lc) return;
  int l = idx & (Lc - 1);
  int rest = idx >> 6;
  int b = rest & (Bc - 1);
  int t = rest >> 8;
  int j = t + l;  // 0..126
  float v = (j < Hc) ? hist[b * Hc + j] : cur[b * Tc + (j - Hc)];
  cs[idx] = v;
}

// cat0[b, 0:512] = meanEnc[b], cat0[b, 512:576] = carStates[0][b]
__global__ void cat0_kernel(const float* __restrict__ meanEnc,
                            const float* __restrict__ cs0,
                            float* __restrict__ cat0) {
  int idx = blockIdx.x * blockDim.x + threadIdx.x;  // over B*576
  if (idx >= Bc * (Ec + Lc)) return;
  int b = idx / (Ec + Lc);
  int k = idx - b * (Ec + Lc);
  cat0[idx] = (k < Ec) ? meanEnc[b * Ec + k] : cs0[b * Lc + (k - Ec)];
}

// -------------------------- per-step kernels -------------------------------
__device__ inline float sigf(float x) { return 1.0f / (1.0f + expf(-x)); }

__global__ void lstm_update_kernel(const float* __restrict__ gates,
                                   float* __restrict__ h, float* __restrict__ c) {
  int idx = blockIdx.x * blockDim.x + threadIdx.x;  // over B*D
  if (idx >= Bc * Dc) return;
  int b = idx / Dc;
  int d = idx - b * Dc;
  const float* g = gates + (size_t)b * G4;
  float gi = sigf(g[d]);
  float gf = sigf(g[Dc + d]);
  float gg = tanhf(g[2 * Dc + d]);
  float go = sigf(g[3 * Dc + d]);
  float cn = gf * c[idx] + gi * gg;
  c[idx] = cn;
  h[idx] = go * tanhf(cn);
}

// p[b] = S * sigmoid( dot(q[b], vp_w) + vp_b ); one wave per b
__global__ __launch_bounds__(256) void p_kernel(const float* __restrict__ q,
                                                const float* __restrict__ vp_w,
                                                const float* __restrict__ vp_b,
                                                float* __restrict__ p) {
  int wave = threadIdx.x >> 5;
  int lane = threadIdx.x & 31;
  int b = blockIdx.x * 8 + wave;
  const float* qb = q + (size_t)b * Dc;
  float s = 0.0f;
  for (int d = lane; d < Dc; d += 32) s += qb[d] * vp_w[d];
#pragma unroll
  for (int off = 16; off; off >>= 1) s += __shfl_xor(s, off, 32);
  if (lane == 0) p[b] = (float)Sc * sigf(s + vp_b[0]);
}

// Fused attention step for one batch b:
//  att[s] = cEnc[b,s,:] . u[b,:]  -> softmax -> * gaussian(p) -> alphas out
//  awe[e] = sum_s alpha[s] * cEnc[b,s,e];  preds = [awe|h] . fc_w + fc_b
__global__ __launch_bounds__(256) void attention_kernel(
    const float* __restrict__ cEnc, const float* __restrict__ u,
    const float* __restrict__ h, const float* __restrict__ p,
    const float* __restrict__ fc_w, const float* __restrict__ fc_b,
    float* __restrict__ preds, float* __restrict__ alphas, int t) {
  __shared__ float s_u[Ec];
  __shared__ float s_att[Sc];
  __shared__ float s_red[32];

  const int b = blockIdx.x;
  const int tid = threadIdx.x;  // 256
  const int wave = tid >> 5;
  const int lane = tid & 31;
  const float* Eb = cEnc + (size_t)b * Sc * Ec;

  s_u[tid] = u[(size_t)b * Ec + tid];
  s_u[tid + 256] = u[(size_t)b * Ec + 256 + tid];
  __syncthreads();

  // Phase A: logits. Each wave handles rows s = wave, wave+8, ...
  for (int s = wave; s < Sc; s += 8) {
    const float* row = Eb + (size_t)s * Ec;
    float acc = 0.0f;
#pragma unroll
    for (int j = 0; j < 4; ++j) {
      int e = lane * 4 + j * 128;
      float4 v = *(const float4*)(row + e);
      acc += v.x * s_u[e] + v.y * s_u[e + 1] + v.z * s_u[e + 2] + v.w * s_u[e + 3];
    }
#pragma unroll
    for (int off = 16; off; off >>= 1) acc += __shfl_xor(acc, off, 32);
    if (lane == 0) s_att[s] = acc;
  }
  __syncthreads();

  // Block max
  float v1 = s_att[tid];
  float v2 = s_att[tid + 256];
  float m = fmaxf(v1, v2);
#pragma unroll
  for (int off = 16; off; off >>= 1) m = fmaxf(m, __shfl_xor(m, off, 32));
  if (lane == 0) s_red[wave] = m;
  __syncthreads();
  m = s_red[0];
#pragma unroll
  for (int i = 1; i < 8; ++i) m = fmaxf(m, s_red[i]);

  // Block sum of exp
  float e1 = expf(v1 - m);
  float e2 = expf(v2 - m);
  float zs = e1 + e2;
#pragma unroll
  for (int off = 16; off; off >>= 1) zs += __shfl_xor(zs, off, 32);
  if (lane == 0) s_red[8 + wave] = zs;
  __syncthreads();
  float Z = 0.0f;
#pragma unroll
  for (int i = 0; i < 8; ++i) Z += s_red[8 + i];
  const float inv = 1.0f / Z;

  // Gaussian window, write alphas, stash weights
  const float pb = p[b];
  float ds1 = (float)tid - pb;
  float ds2 = (float)(tid + 256) - pb;
  float w1 = e1 * inv * expf(-ds1 * ds1 * 0.125f);
  float w2 = e2 * inv * expf(-ds2 * ds2 * 0.125f);
  float* arow = alphas + ((size_t)b * Tc + t) * Sc;
  arow[tid] = w1;
  arow[tid + 256] = w2;
  __syncthreads();  // everyone done reading s_att / s_red
  s_att[tid] = w1;
  s_att[tid + 256] = w2;
  __syncthreads();

  // Phase B: awe (thread owns columns tid and tid+256)
  float a0 = 0.0f, a1 = 0.0f;
  for (int s = 0; s < Sc; ++s) {
    float w = s_att[s];
    const float* row = Eb + (size_t)s * Ec;
    a0 += w * row[tid];
    a1 += w * row[tid + 256];
  }

  // preds = [awe | h] . fc_w + fc_b
  const float* hb = h + (size_t)b * Dc;
  float pr = a0 * fc_w[tid] + a1 * fc_w[tid + 256] +
             hb[tid] * fc_w[Ec + tid] + hb[tid + 256] * fc_w[Ec + 256 + tid];
#pragma unroll
  for (int off = 16; off; off >>= 1) pr += __shfl_xor(pr, off, 32);
  if (lane == 0) s_red[16 + wave] = pr;
  __syncthreads();
  if (tid == 0) {
    float tot = 0.0f;
#pragma unroll
    for (int i = 0; i < 8; ++i) tot += s_red[16 + i];
    preds[(size_t)b * Tc + t] = tot + fc_b[0];
  }
}

// ---------------------------------------------------------------------------
extern "C" void kernel_launch(void* const* d_in, const int* in_sizes, int n_in,
                              void* d_out, int out_size, void* d_ws,
                              size_t ws_size, hipStream_t stream) {
  (void)in_sizes; (void)n_in; (void)out_size; (void)ws_size;
  const float* cEnc = (const float*)d_in[0];
  const float* cur = (const float*)d_in[1];
  const float* hist = (const float*)d_in[2];
  // d_in[3], d_in[4] (accel) unused; d_in[5] decodeLength == Tc
  const float* Wa_w = (const float*)d_in[6];
  // Wa_b cancels in softmax
  const float* Wp_w = (const float*)d_in[8];
  const float* Wp_b = (const float*)d_in[9];
  const float* vp_w = (const float*)d_in[10];
  const float* vp_b = (const float*)d_in[11];
  const float* ih_w = (const float*)d_in[12];
  const float* ih_b = (const float*)d_in[13];
  const float* ic_w = (const float*)d_in[14];
  const float* ic_b = (const float*)d_in[15];
  const float* W_ih = (const float*)d_in[16];
  const float* W_hh = (const float*)d_in[17];
  const float* b_ih = (const float*)d_in[18];
  const float* b_hh = (const float*)d_in[19];
  const float* fc_w = (const float*)d_in[20];
  const float* fc_b = (const float*)d_in[21];

  float* out = (float*)d_out;
  float* preds = out;                       // (B, T, 1)
  float* alphas = out + (size_t)Bc * Tc;    // (B, T, S)

  char* ws = (char*)d_ws;
  float* meanEnc = (float*)ws; ws += (size_t)Bc * Ec * 4;
  float* cs      = (float*)ws; ws += (size_t)Tc * Bc * Lc * 4;
  float* cat0    = (float*)ws; ws += (size_t)Bc * (Ec + Lc) * 4;
  float* hbuf    = (float*)ws; ws += (size_t)Bc * Dc * 4;
  float* cbuf    = (float*)ws; ws += (size_t)Bc * Dc * 4;
  float* gates   = (float*)ws; ws += (size_t)Bc * G4 * 4;
  float* qbuf    = (float*)ws; ws += (size_t)Bc * Dc * 4;
  float* ubuf    = (float*)ws; ws += (size_t)Bc * Ec * 4;
  float* pbuf    = (float*)ws; ws += (size_t)Bc * 4;

  // ---- setup ----
  mean_enc_kernel<<<(Bc * Ec + 255) / 256, 256, 0, stream>>>(cEnc, meanEnc);
  car_states_kernel<<<(Tc * Bc * Lc + 255) / 256, 256, 0, stream>>>(hist, cur, cs);
  cat0_kernel<<<(Bc * (Ec + Lc) + 255) / 256, 256, 0, stream>>>(meanEnc, cs, cat0);

  // h0 / c0 : (B,576) @ (576,512) + bias
  gemm_wmma_kernel<false, 0><<<dim3(Bc / 16, Dc / 128), 256, 0, stream>>>(
      cat0, Ec + Lc, ih_w, Dc, ih_b, hbuf, Dc, Ec + Lc);
  gemm_wmma_kernel<false, 0><<<dim3(Bc / 16, Dc / 128), 256, 0, stream>>>(
      cat0, Ec + Lc, ic_w, Dc, ic_b, cbuf, Dc, Ec + Lc);

  // ---- decode loop ----
  for (int t = 0; t < Tc; ++t) {
    const float* xt = cs + (size_t)t * Bc * Lc;
    gates_wmma_kernel<<<dim3(Bc / 16, G4 / 128), 256, 0, stream>>>(
        xt, hbuf, W_ih, W_hh, b_ih, b_hh, gates);
    lstm_update_kernel<<<(Bc * Dc + 255) / 256, 256, 0, stream>>>(gates, hbuf, cbuf);
    // q = tanh(h @ Wp + bp)
    gemm_wmma_kernel<false, 1><<<dim3(Bc / 16, Dc / 128), 256, 0, stream>>>(
        hbuf, Dc, Wp_w, Dc, Wp_b, qbuf, Dc, Dc);
    // u = h @ Wa_w^T   (u[b,e] = sum_d Wa_w[e,d] h[b,d])
    gemm_wmma_kernel<true, 0><<<dim3(Bc / 16, Ec / 128), 256, 0, stream>>>(
        hbuf, Dc, Wa_w, Dc, nullptr, ubuf, Ec, Dc);
    p_kernel<<<Bc / 8, 256, 0, stream>>>(qbuf, vp_w, vp_b, pbuf);
    attention_kernel<<<Bc, 256, 0, stream>>>(cEnc, ubuf, hbuf, pbuf, fc_w, fc_b,
                                             preds, alphas, t);
  }
}